// HDLoss_29712583753732
// MI455X (gfx1250) — compile-verified
//
#include <hip/hip_runtime.h>
#include <hip/hip_bf16.h>

// ---------------------------------------------------------------------------
// HD loss: softmax -> binary masks -> exact squared EDT (3 separable O(n^2)
// 1-D min-plus passes, matching the reference's float32 arithmetic) ->
// mean((p1-y1)^2 * (edtP+edtG)).
//
// CDNA5 paths: global_load_async_to_lds_b128 + s_wait_asynccnt for the
// strided y/z EDT passes; v_wmma_f32_16x16x4_f32 for wave reductions.
// ---------------------------------------------------------------------------

#define EDT_INF 1e10f
#define VOL     2097152      // 128^3
#define NVOX    8388608      // 4 * 128^3  (loss is over channel 1 only)

typedef float v2f __attribute__((ext_vector_type(2)));
typedef float v8f __attribute__((ext_vector_type(8)));

// Wave-level sum of one float per lane using V_WMMA_F32_16X16X4_F32.
// A[m][k] carries the 32 lane partials (k=0 lanes 0-15, k=2 lanes 16-31),
// B = ones => D[m][n] = partial(m) + partial(m+16) for every n.
// Sum of a lane's 8 accumulator VGPRs = half-sum (m=0..7 or m=8..15);
// one shfl_xor(16) completes the total (same value in every lane).
__device__ inline float wave_sum_wmma(float partial) {
    v2f a;    a[0] = partial; a[1] = 0.0f;
    v2f ones; ones[0] = 1.0f; ones[1] = 1.0f;
    v8f c = {};
    c = __builtin_amdgcn_wmma_f32_16x16x4_f32(
            false, a, false, ones, (short)0, c, false, false);
    float h = c[0] + c[1] + c[2] + c[3] + c[4] + c[5] + c[6] + c[7];
    return h + __shfl_xor(h, 16, 32);
}

// ---------------------------------------------------------------------------
// Kernel 1: build both binary "f" fields and do the x-axis (contiguous) pass.
// One 128-thread block per (b, d0, d1) line; line data lives in LDS.
// vols[0..3] = P volumes, vols[4..7] = G volumes.
// ---------------------------------------------------------------------------
__global__ void __launch_bounds__(128)
edt_init_x_pass(const float* __restrict__ net, const int* __restrict__ gt,
                float* __restrict__ vols) {
    __shared__ float sP[128];
    __shared__ float sG[128];

    const int x    = threadIdx.x;
    const int line = blockIdx.x;          // ((b*128)+d0)*128 + d1
    const int b    = line >> 14;
    const int rem  = line & 16383;

    const size_t voff  = ((size_t)rem << 7) + (size_t)x;   // index in 128^3
    const size_t nbase = (size_t)b * 2 * VOL;

    const float n0 = net[nbase + voff];
    const float n1 = net[nbase + VOL + voff];
    const int   g  = gt[(size_t)b * VOL + voff];

    // p1 > 0.5  <=>  n1 > n0  (sigmoid(0) == 0.5, strict compare)
    sP[x] = (n1 > n0) ? EDT_INF : 0.0f;
    sG[x] = (g == 1)  ? EDT_INF : 0.0f;
    __syncthreads();

    float dP = EDT_INF, dG = EDT_INF;
    const float fx = (float)x;
    for (int j = 0; j < 128; ++j) {
        const float t  = fx - (float)j;
        const float tt = t * t;                 // exact (|t| <= 127)
        dP = fminf(dP, tt + sP[j]);
        dG = fminf(dG, tt + sG[j]);
    }

    vols[(size_t)b * VOL + voff]                     = dP;  // P volume b
    vols[(size_t)(b + 4) * VOL + voff]               = dG;  // G volume b
}

// ---------------------------------------------------------------------------
// Kernel 2: strided EDT pass (axis with element stride `line_stride`),
// in place on all 8 volumes.  Each 256-thread block owns a 128(row=j) x
// 32(col) tile: 128 rows of 128 contiguous bytes are staged into LDS with
// global_load_async_to_lds_b128 (one instr moves 4 rows per wave), then
// each wave computes 16 transform outputs for its 32 columns.
//   y-pass: outer_stride = 16384, line_stride = 128
//   z-pass: outer_stride = 128,   line_stride = 16384
// ---------------------------------------------------------------------------
__global__ void __launch_bounds__(256)
edt_strided_pass(float* __restrict__ vols, int outer_stride, int line_stride) {
    __shared__ float tile[128 * 32];      // 16 KB, [row][col], conflict-free

    const int tid  = threadIdx.x;
    const int lane = tid & 31;            // column within tile
    const int wave = tid >> 5;            // 0..7

    const int bx    = blockIdx.x;         // ((vol*128)+outer)*4 + xb
    const int xb    = bx & 3;
    const int outer = (bx >> 2) & 127;
    const int vol   = bx >> 9;            // 0..7

    float* base = vols + (size_t)vol * VOL
                       + (size_t)outer * (size_t)outer_stride
                       + (size_t)xb * 32;

    // --- async stage: 128 rows x 128B.  Per wave-instruction: 32 lanes x
    // 16B = 4 rows; each of 8 waves issues 4 instructions (rows w*16..w*16+15).
    const uint32_t tile_lds = (uint32_t)(size_t)(&tile[0]);  // low 32b = LDS off
    const int rsub = lane >> 3;          // 0..3 : row within 4-row group
    const int xq   = lane & 7;           // 0..7 : 16B chunk within row
    #pragma unroll
    for (int it = 0; it < 4; ++it) {
        const int row = wave * 16 + it * 4 + rsub;
        const float*   gaddr = base + (size_t)row * (size_t)line_stride + xq * 4;
        const uint32_t laddr = tile_lds + (uint32_t)(row * 128 + xq * 16);
        asm volatile("global_load_async_to_lds_b128 %0, %1, off"
                     :: "v"(laddr), "v"(gaddr) : "memory");
    }
    asm volatile("s_wait_asynccnt 0" ::: "memory");
    __syncthreads();

    // --- min-plus transform: wave w computes outputs i = w + 8k, k=0..15,
    // lanes are the 32 columns.  One LDS read feeds 16 fma+min pairs.
    float d[16];
    #pragma unroll
    for (int k = 0; k < 16; ++k) d[k] = EDT_INF;

    for (int j = 0; j < 128; ++j) {
        const float f  = tile[j * 32 + lane];
        const float fj = (float)j;
        #pragma unroll
        for (int k = 0; k < 16; ++k) {
            const float t = (float)(wave + 8 * k) - fj;   // hoisted cvt
            d[k] = fminf(d[k], __builtin_fmaf(t, t, f));  // exact t*t + f
        }
    }

    #pragma unroll
    for (int k = 0; k < 16; ++k) {
        const int i = wave + 8 * k;
        base[(size_t)i * (size_t)line_stride + lane] = d[k];   // coalesced
    }
}

// ---------------------------------------------------------------------------
// Kernel 3: fused loss + deterministic block partials.
// loss_v = (sigmoid(n1-n0) - [gt==1])^2 * (edtP + edtG), scaled by 1/N.
// Wave totals via WMMA; block total written to partials[blockIdx].
// ---------------------------------------------------------------------------
__global__ void __launch_bounds__(256)
loss_partial(const float* __restrict__ net, const int* __restrict__ gt,
             const float* __restrict__ vols, float* __restrict__ partials) {
    const float* volP = vols;
    const float* volG = vols + (size_t)4 * VOL;

    float acc = 0.0f;
    const size_t stride = (size_t)gridDim.x * blockDim.x;
    for (size_t v = (size_t)blockIdx.x * blockDim.x + threadIdx.x;
         v < (size_t)NVOX; v += stride) {
        const size_t b    = v >> 21;
        const size_t voff = v & (VOL - 1);
        const size_t nidx = b * 2 * VOL + voff;
        const float n0 = net[nidx];
        const float n1 = net[nidx + VOL];
        const float p1 = 1.0f / (1.0f + expf(n0 - n1));   // softmax ch-1
        const float y1 = (gt[v] == 1) ? 1.0f : 0.0f;
        const float e  = p1 - y1;
        acc += e * e * (volP[v] + volG[v]);
    }
    acc *= (1.0f / (float)NVOX);

    const float wtot = wave_sum_wmma(acc);

    __shared__ float wsum[8];
    const int lane = threadIdx.x & 31, wave = threadIdx.x >> 5;
    if (lane == 0) wsum[wave] = wtot;
    __syncthreads();
    if (threadIdx.x == 0) {
        float s = 0.0f;
        #pragma unroll
        for (int w = 0; w < 8; ++w) s += wsum[w];
        partials[blockIdx.x] = s;
    }
}

// ---------------------------------------------------------------------------
// Kernel 4: single-block deterministic final reduction -> d_out[0].
// ---------------------------------------------------------------------------
__global__ void __launch_bounds__(256)
final_reduce(const float* __restrict__ partials, int n, float* __restrict__ out) {
    float acc = 0.0f;
    for (int i = threadIdx.x; i < n; i += blockDim.x) acc += partials[i];

    const float wtot = wave_sum_wmma(acc);

    __shared__ float wsum[8];
    const int lane = threadIdx.x & 31, wave = threadIdx.x >> 5;
    if (lane == 0) wsum[wave] = wtot;
    __syncthreads();
    if (threadIdx.x == 0) {
        float s = 0.0f;
        #pragma unroll
        for (int w = 0; w < 8; ++w) s += wsum[w];
        out[0] = s;
    }
}

// ---------------------------------------------------------------------------
extern "C" void kernel_launch(void* const* d_in, const int* in_sizes, int n_in,
                              void* d_out, int out_size, void* d_ws, size_t ws_size,
                              hipStream_t stream) {
    (void)in_sizes; (void)n_in; (void)out_size; (void)ws_size;

    const float* net = (const float*)d_in[0];   // (4,2,128,128,128) f32
    const int*   gt  = (const int*)d_in[1];     // (4,1,128,128,128) int
    float*       out = (float*)d_out;           // scalar

    // ws: 8 x 128^3 f32 EDT volumes (64 MB), then 1024 block partials.
    float* vols     = (float*)d_ws;
    float* partials = vols + (size_t)8 * VOL;

    const int NBLK = 1024;

    // 1) masks + x-axis pass (contiguous lines)
    edt_init_x_pass<<<4 * 128 * 128, 128, 0, stream>>>(net, gt, vols);

    // 2) y-axis pass (stride 128), then z-axis pass (stride 128*128),
    //    in place over all 8 volumes; async-LDS staged tiles.
    edt_strided_pass<<<8 * 128 * 4, 256, 0, stream>>>(vols, 16384, 128);
    edt_strided_pass<<<8 * 128 * 4, 256, 0, stream>>>(vols, 128, 16384);

    // 3) fused loss + partials, 4) deterministic final sum
    loss_partial<<<NBLK, 256, 0, stream>>>(net, gt, vols, partials);
    final_reduce<<<1, 256, 0, stream>>>(partials, NBLK, out);
}